// RGCNDecoder_30013231464960
// MI455X (gfx1250) — compile-verified
//
#include <hip/hip_runtime.h>
#include <math.h>

// ---------------- problem constants ----------------
constexpr int N = 50000, E = 800000, D = 128, R = 16, L = 5;
constexpr int OBJ_DIM = 96, ATTR_DIM = 32, HID = 512, NANGLE = 24;
constexpr int ET = 64;  // edges (or rows) per block in the D=128 GEMM kernels

static_assert(N % 16 == 0, "N tile");
static_assert(E % ET == 0, "E tile");

typedef __attribute__((ext_vector_type(2))) float v2f;
typedef __attribute__((ext_vector_type(8))) float v8f;
typedef int v4i __attribute__((vector_size(16)));

__device__ __forceinline__ v8f wmma_f32_k4(v2f a, v2f b, v8f c) {
  // D = A(16x4, f32) * B(4x16, f32) + C(16x16, f32)
  return __builtin_amdgcn_wmma_f32_16x16x4_f32(
      /*neg_a=*/false, a, /*neg_b=*/false, b,
      /*c_mod=*/(short)0, c, /*reuse_a=*/false, /*reuse_b=*/false);
}

// ---- CDNA5 async global->LDS copy (16B per lane), ASYNCcnt tracked --------
#if defined(__gfx1250__) && __has_builtin(__builtin_amdgcn_global_load_async_to_lds_b128)
#define HAVE_ASYNC_LDS 1
#else
#define HAVE_ASYNC_LDS 0
#endif

__device__ __forceinline__ void stage16(const float* __restrict__ g,
                                        float* __restrict__ l) {
#if HAVE_ASYNC_LDS
  __builtin_amdgcn_global_load_async_to_lds_b128(
      (__attribute__((address_space(1))) v4i*)g,
      (__attribute__((address_space(3))) v4i*)l, 0, 0);
#else
  *(float4*)l = *(const float4*)g;
#endif
}

__device__ __forceinline__ void stage_wait() {
#if HAVE_ASYNC_LDS
#if __has_builtin(__builtin_amdgcn_s_wait_asynccnt)
  __builtin_amdgcn_s_wait_asynccnt(0);
#else
  asm volatile("s_wait_asynccnt 0x0" ::: "memory");
#endif
#endif
}

// ---------------- embedding / attr cache ----------------
__global__ __launch_bounds__(256) void embed_kernel(
    const int* __restrict__ objs, const int* __restrict__ attrs,
    const float* __restrict__ obj_emb, const float* __restrict__ attr_emb,
    float* __restrict__ x0, float* __restrict__ attrv) {
  int idx = blockIdx.x * 256 + threadIdx.x;
  if (idx >= N * D) return;
  int n = idx >> 7, d = idx & 127;
  float v;
  if (d < OBJ_DIM) {
    v = obj_emb[objs[n] * OBJ_DIM + d];
  } else {
    v = attr_emb[attrs[n] * ATTR_DIM + (d - OBJ_DIM)];
    attrv[n * ATTR_DIM + (d - OBJ_DIM)] = v;
  }
  x0[idx] = v;
}

// ---------------- (dest,rel) degree + relation histogram ----------------
__global__ __launch_bounds__(256) void count_kernel(
    const int* __restrict__ triples, int* __restrict__ cnt, int* __restrict__ hist) {
  int e = blockIdx.x * 256 + threadIdx.x;
  if (e >= E) return;
  int p = triples[e * 3 + 1];
  int o = triples[e * 3 + 2];
  atomicAdd(&cnt[o * R + p], 1);
  atomicAdd(&hist[p], 1);
}

__global__ void prefix_kernel(const int* __restrict__ hist, int* __restrict__ pos) {
  if (threadIdx.x == 0) {
    int acc = 0;
    for (int i = 0; i < R; ++i) { pos[i] = acc; acc += hist[i]; }
  }
}

// counting-sort edges by relation; precompute per-edge 1/deg
__global__ __launch_bounds__(256) void sort_kernel(
    const int* __restrict__ triples, const int* __restrict__ cnt,
    int* __restrict__ pos, int* __restrict__ es, int* __restrict__ eo,
    int* __restrict__ ep, float* __restrict__ einv) {
  int e = blockIdx.x * 256 + threadIdx.x;
  if (e >= E) return;
  int s = triples[e * 3 + 0];
  int p = triples[e * 3 + 1];
  int o = triples[e * 3 + 2];
  int slot = atomicAdd(&pos[p], 1);
  es[slot] = s; eo[slot] = o; ep[slot] = p;
  einv[slot] = 1.0f / fmaxf((float)cnt[o * R + p], 1.0f);
}

// ---------------- edge message pass: agg[o] += (x[s] @ Wrel[p]) / deg ------
// 64 sorted edges per block (4 row tiles of 16). B fragments loaded once per
// K-step and reused across the 4 row tiles. Relation uniform except <=15
// boundary blocks which take the scalar fallback.
__global__ __launch_bounds__(256) void rgcn_edge_kernel(
    const float* __restrict__ xin, float* __restrict__ agg,
    const float* __restrict__ Wrel_j,
    const int* __restrict__ es, const int* __restrict__ eo,
    const int* __restrict__ ep, const float* __restrict__ einv) {
  __shared__ float xs[ET * 128];          // 32 KB gathered source rows
  __shared__ int esL[ET], eoL[ET];
  __shared__ float invL[ET];
  __shared__ int pends[2];
  const int t = threadIdx.x;
  const int base = blockIdx.x * ET;
  if (t < ET) {
    esL[t] = es[base + t]; eoL[t] = eo[base + t]; invL[t] = einv[base + t];
  }
  if (t == 0) pends[0] = ep[base];
  if (t == 1) pends[1] = ep[base + ET - 1];
  __syncthreads();
  // gather ET rows (ETx128 f32) into LDS, 16B per lane, async path on CDNA5
  for (int idx = t; idx < ET * 32; idx += 256) {
    int row = idx >> 5, c4 = idx & 31;
    stage16(xin + (size_t)esL[row] * D + c4 * 4, &xs[row * 128 + c4 * 4]);
  }
  stage_wait();
  __syncthreads();

  if (pends[0] == pends[1]) {  // uniform relation -> WMMA path
    const float* W = Wrel_j + (size_t)pends[0] * D * D;  // [k][f] row-major
    const int lane = t & 31, wave = t >> 5;
    const int n = lane & 15;            // A row / B col / D col within tile
    const int kh = (lane >> 4) << 1;    // K half select (0 or 2)
    const int col0 = wave * 16;         // this wave's 16 output columns
    const float* wp = W + (size_t)kh * D + col0 + n;
    v8f c0 = {0.f,0.f,0.f,0.f,0.f,0.f,0.f,0.f};
    v8f c1 = c0, c2 = c0, c3 = c0;
    for (int k = 0; k < D; k += 4) {
      v2f b;
      b.x = wp[(size_t)k * D];
      b.y = wp[(size_t)(k + 1) * D];
      v2f a;
      a.x = xs[(0 * 16 + n) * 128 + k + kh]; a.y = xs[(0 * 16 + n) * 128 + k + kh + 1];
      c0 = wmma_f32_k4(a, b, c0);
      a.x = xs[(1 * 16 + n) * 128 + k + kh]; a.y = xs[(1 * 16 + n) * 128 + k + kh + 1];
      c1 = wmma_f32_k4(a, b, c1);
      a.x = xs[(2 * 16 + n) * 128 + k + kh]; a.y = xs[(2 * 16 + n) * 128 + k + kh + 1];
      c2 = wmma_f32_k4(a, b, c2);
      a.x = xs[(3 * 16 + n) * 128 + k + kh]; a.y = xs[(3 * 16 + n) * 128 + k + kh + 1];
      c3 = wmma_f32_k4(a, b, c3);
    }
    const int rbase = (lane < 16) ? 0 : 8;
#pragma unroll
    for (int v = 0; v < 8; ++v) {
      int r0 = v + rbase;
      atomicAdd(&agg[(size_t)eoL[r0] * D + col0 + n],      c0[v] * invL[r0]);
      atomicAdd(&agg[(size_t)eoL[16 + r0] * D + col0 + n], c1[v] * invL[16 + r0]);
      atomicAdd(&agg[(size_t)eoL[32 + r0] * D + col0 + n], c2[v] * invL[32 + r0]);
      atomicAdd(&agg[(size_t)eoL[48 + r0] * D + col0 + n], c3[v] * invL[48 + r0]);
    }
  } else {  // relation boundary block: scalar fallback (4 threads per edge)
    const int i = t & 63;
    const int f0 = (t >> 6) * 32;
    const float* Wp = Wrel_j + (size_t)ep[base + i] * D * D;
    for (int f = f0; f < f0 + 32; ++f) {
      float acc = 0.f;
      for (int k = 0; k < D; ++k) acc += xs[i * 128 + k] * Wp[(size_t)k * D + f];
      atomicAdd(&agg[(size_t)eoL[i] * D + f], acc * invL[i]);
    }
  }
}

// ---------------- root transform: xio = relu(x @ Wroot + agg + b) ----------
// 64 node rows per block, B fragments shared across the 4 row tiles.
__global__ __launch_bounds__(256) void rgcn_root_kernel(
    const float* __restrict__ xin, float* __restrict__ xio,
    const float* __restrict__ Wroot_j, const float* __restrict__ b_j) {
  __shared__ float xs[ET * 128];
  const int t = threadIdx.x;
  const int base = blockIdx.x * ET;
  for (int idx = t; idx < ET * 32; idx += 256) {
    int row = idx >> 5, c4 = idx & 31;
    int node = base + row; if (node >= N) node = N - 1;  // tail clamp
    stage16(xin + (size_t)node * D + c4 * 4, &xs[row * 128 + c4 * 4]);
  }
  stage_wait();
  __syncthreads();
  const int lane = t & 31, wave = t >> 5;
  const int n = lane & 15, kh = (lane >> 4) << 1;
  const int col0 = wave * 16;
  const float* wp = Wroot_j + (size_t)kh * D + col0 + n;
  v8f c0 = {0.f,0.f,0.f,0.f,0.f,0.f,0.f,0.f};
  v8f c1 = c0, c2 = c0, c3 = c0;
  for (int k = 0; k < D; k += 4) {
    v2f b;
    b.x = wp[(size_t)k * D];
    b.y = wp[(size_t)(k + 1) * D];
    v2f a;
    a.x = xs[(0 * 16 + n) * 128 + k + kh]; a.y = xs[(0 * 16 + n) * 128 + k + kh + 1];
    c0 = wmma_f32_k4(a, b, c0);
    a.x = xs[(1 * 16 + n) * 128 + k + kh]; a.y = xs[(1 * 16 + n) * 128 + k + kh + 1];
    c1 = wmma_f32_k4(a, b, c1);
    a.x = xs[(2 * 16 + n) * 128 + k + kh]; a.y = xs[(2 * 16 + n) * 128 + k + kh + 1];
    c2 = wmma_f32_k4(a, b, c2);
    a.x = xs[(3 * 16 + n) * 128 + k + kh]; a.y = xs[(3 * 16 + n) * 128 + k + kh + 1];
    c3 = wmma_f32_k4(a, b, c3);
  }
  const float bias = b_j[col0 + n];
  const int rbase = (lane < 16) ? 0 : 8;
#pragma unroll
  for (int v = 0; v < 8; ++v) {
#pragma unroll
    for (int rt = 0; rt < 4; ++rt) {
      int node = base + rt * 16 + v + rbase;
      if (node < N) {
        size_t o = (size_t)node * D + col0 + n;
        float acc = (rt == 0 ? c0[v] : rt == 1 ? c1[v] : rt == 2 ? c2[v] : c3[v]);
        xio[o] = fmaxf(acc + xio[o] + bias, 0.f);
      }
    }
  }
}

// ---------------- fused box head: relu([x,z,attr] @ W1 + b1) @ W2 + b2 -----
__global__ __launch_bounds__(256) void box_kernel(
    const float* __restrict__ x5, const float* __restrict__ z,
    const float* __restrict__ attrv,
    const float* __restrict__ W1, const float* __restrict__ b1,
    const float* __restrict__ W2, const float* __restrict__ b2,
    float* __restrict__ out) {
  constexpr int F = 2 * D + ATTR_DIM;  // 288
  __shared__ float feat[16 * F];
  __shared__ float hid[16 * HID];
  const int t = threadIdx.x, base = blockIdx.x * 16;
  // 72 16B chunks per node row: 32 from x5, 32 from z, 8 from attrv
  for (int idx = t; idx < 16 * 72; idx += 256) {
    int row = idx / 72, ch = idx % 72;
    int node = base + row;
    const float* src;
    if (ch < 32)      src = x5 + (size_t)node * D + ch * 4;
    else if (ch < 64) src = z + (size_t)node * D + (ch - 32) * 4;
    else              src = attrv + (size_t)node * ATTR_DIM + (ch - 64) * 4;
    stage16(src, &feat[row * F + ch * 4]);
  }
  stage_wait();
  __syncthreads();
  const int lane = t & 31, wave = t >> 5;
  const int n = lane & 15, kh = (lane >> 4) << 1;
  for (int tt = 0; tt < 4; ++tt) {  // 8 waves * 4 tiles = 512 hidden cols
    const int col0 = wave * 64 + tt * 16;
    v8f c = {0.f,0.f,0.f,0.f,0.f,0.f,0.f,0.f};
    for (int k = 0; k < F; k += 4) {
      v2f a, b;
      a.x = feat[n * F + k + kh];
      a.y = feat[n * F + k + kh + 1];
      b.x = W1[(size_t)(k + kh) * HID + col0 + n];
      b.y = W1[(size_t)(k + kh + 1) * HID + col0 + n];
      c = wmma_f32_k4(a, b, c);
    }
    const float bias = b1[col0 + n];
#pragma unroll
    for (int v = 0; v < 8; ++v) {
      int row = v + ((lane < 16) ? 0 : 8);
      hid[row * HID + col0 + n] = fmaxf(c[v] + bias, 0.f);
    }
  }
  __syncthreads();
  if (t < 16 * 6) {
    int row = t / 6, cc = t % 6;
    float acc = b2[cc];
    for (int h = 0; h < HID; ++h) acc += hid[row * HID + h] * W2[h * 6 + cc];
    out[(size_t)(base + row) * 6 + cc] = acc;
  }
}

// ---------------- fused angle head + log_softmax ---------------------------
__global__ __launch_bounds__(256) void ang_kernel(
    const float* __restrict__ x5, const float* __restrict__ z,
    const float* __restrict__ W1, const float* __restrict__ b1,
    const float* __restrict__ W2, const float* __restrict__ b2,
    float* __restrict__ outA) {
  constexpr int F = 2 * D;  // 256
  __shared__ float feat[16 * F];
  __shared__ float hid[16 * HID];
  __shared__ float logits[16 * NANGLE];
  const int t = threadIdx.x, base = blockIdx.x * 16;
  for (int idx = t; idx < 16 * 64; idx += 256) {  // 64 16B chunks per node
    int row = idx >> 6, ch = idx & 63;
    int node = base + row;
    const float* src = (ch < 32) ? x5 + (size_t)node * D + ch * 4
                                 : z + (size_t)node * D + (ch - 32) * 4;
    stage16(src, &feat[row * F + ch * 4]);
  }
  stage_wait();
  __syncthreads();
  const int lane = t & 31, wave = t >> 5;
  const int n = lane & 15, kh = (lane >> 4) << 1;
  for (int tt = 0; tt < 4; ++tt) {
    const int col0 = wave * 64 + tt * 16;
    v8f c = {0.f,0.f,0.f,0.f,0.f,0.f,0.f,0.f};
    for (int k = 0; k < F; k += 4) {
      v2f a, b;
      a.x = feat[n * F + k + kh];
      a.y = feat[n * F + k + kh + 1];
      b.x = W1[(size_t)(k + kh) * HID + col0 + n];
      b.y = W1[(size_t)(k + kh + 1) * HID + col0 + n];
      c = wmma_f32_k4(a, b, c);
    }
    const float bias = b1[col0 + n];
#pragma unroll
    for (int v = 0; v < 8; ++v) {
      int row = v + ((lane < 16) ? 0 : 8);
      hid[row * HID + col0 + n] = fmaxf(c[v] + bias, 0.f);
    }
  }
  __syncthreads();
  for (int idx = t; idx < 16 * NANGLE; idx += 256) {
    int row = idx / NANGLE, cc = idx % NANGLE;
    float acc = b2[cc];
    for (int h = 0; h < HID; ++h) acc += hid[row * HID + h] * W2[h * NANGLE + cc];
    logits[idx] = acc;
  }
  __syncthreads();
  if (t < 16) {
    float mx = -1e30f;
    for (int cc = 0; cc < NANGLE; ++cc) mx = fmaxf(mx, logits[t * NANGLE + cc]);
    float s = 0.f;
    for (int cc = 0; cc < NANGLE; ++cc) s += expf(logits[t * NANGLE + cc] - mx);
    float lse = mx + logf(s);
    for (int cc = 0; cc < NANGLE; ++cc)
      outA[(size_t)(base + t) * NANGLE + cc] = logits[t * NANGLE + cc] - lse;
  }
}

// ---------------- host orchestration ----------------
extern "C" void kernel_launch(void* const* d_in, const int* in_sizes, int n_in,
                              void* d_out, int out_size, void* d_ws, size_t ws_size,
                              hipStream_t stream) {
  (void)in_sizes; (void)n_in; (void)out_size; (void)ws_size;
  const float* z        = (const float*)d_in[0];
  const int*   objs     = (const int*)d_in[1];
  const int*   triples  = (const int*)d_in[2];
  const int*   attrs    = (const int*)d_in[3];
  const float* obj_emb  = (const float*)d_in[4];
  const float* attr_emb = (const float*)d_in[5];
  const float* Wrel     = (const float*)d_in[6];
  const float* Wroot    = (const float*)d_in[7];
  const float* bconv    = (const float*)d_in[8];
  const float* box_W1   = (const float*)d_in[9];
  const float* box_b1   = (const float*)d_in[10];
  const float* box_W2   = (const float*)d_in[11];
  const float* box_b2   = (const float*)d_in[12];
  const float* ang_W1   = (const float*)d_in[13];
  const float* ang_b1   = (const float*)d_in[14];
  const float* ang_W2   = (const float*)d_in[15];
  const float* ang_b2   = (const float*)d_in[16];
  float* out = (float*)d_out;

  // workspace layout (~74 MB)
  char* ws = (char*)d_ws;
  float* xA   = (float*)ws;  ws += (size_t)N * D * 4;
  float* xB   = (float*)ws;  ws += (size_t)N * D * 4;
  int*   es   = (int*)ws;    ws += (size_t)E * 4;
  int*   eo   = (int*)ws;    ws += (size_t)E * 4;
  int*   ep   = (int*)ws;    ws += (size_t)E * 4;
  float* einv = (float*)ws;  ws += (size_t)E * 4;
  int*   cnt  = (int*)ws;    ws += (size_t)N * R * 4;
  int*   hist = (int*)ws;    ws += 16 * 4;
  int*   pos  = (int*)ws;    ws += 16 * 4;
  float* attrv = (float*)ws; ws += (size_t)N * ATTR_DIM * 4;

  // per-call re-init (graph-replay safe)
  (void)hipMemsetAsync(cnt, 0, (size_t)N * R * 4, stream);
  (void)hipMemsetAsync(hist, 0, 32 * 4, stream);  // hist + pos

  embed_kernel<<<(N * D) / 256, 256, 0, stream>>>(objs, attrs, obj_emb, attr_emb, xA, attrv);
  count_kernel<<<(E + 255) / 256, 256, 0, stream>>>(triples, cnt, hist);
  prefix_kernel<<<1, 32, 0, stream>>>(hist, pos);
  sort_kernel<<<(E + 255) / 256, 256, 0, stream>>>(triples, cnt, pos, es, eo, ep, einv);

  float* cur = xA;
  float* nxt = xB;
  for (int j = 0; j < L; ++j) {
    (void)hipMemsetAsync(nxt, 0, (size_t)N * D * 4, stream);  // agg accumulator
    rgcn_edge_kernel<<<E / ET, 256, 0, stream>>>(
        cur, nxt, Wrel + (size_t)j * R * D * D, es, eo, ep, einv);
    rgcn_root_kernel<<<(N + ET - 1) / ET, 256, 0, stream>>>(
        cur, nxt, Wroot + (size_t)j * D * D, bconv + (size_t)j * D);
    float* tmp = cur; cur = nxt; nxt = tmp;
  }

  box_kernel<<<N / 16, 256, 0, stream>>>(cur, z, attrv, box_W1, box_b1, box_W2, box_b2, out);
  ang_kernel<<<N / 16, 256, 0, stream>>>(cur, z, ang_W1, ang_b1, ang_W2, ang_b2,
                                         out + (size_t)N * 6);
}